// AttnDecoderRNN_2542620639277
// MI455X (gfx1250) — compile-verified
//
#include <hip/hip_runtime.h>
#include <hip/hip_bf16.h>

#define B_ 64
#define S_ 60
#define T_ 60
#define H_ 1024
#define V_ 32000
#define SOS_ 1

typedef __attribute__((ext_vector_type(2))) float v2f;
typedef __attribute__((ext_vector_type(8))) float v8f;

// ---------------------------------------------------------------------------
// Generic fp32 WMMA GEMM:  C[M x N] = A[M x K] * W[N x K]^T + bias[N]
// Row-major A, W. C row stride = ldc. Requires M%64==0, N%128==0, K%16==0.
// Block: 256 threads (8 waves). Block tile: 64x128. Wave tile: 16x64
// (four 16x16 WMMA accumulators). V_WMMA_F32_16X16X4_F32, K chunked by 16
// through LDS (stride-17 padding -> conflict-free fragment reads).
// ---------------------------------------------------------------------------
__global__ __launch_bounds__(256) void gemm_atb(
    const float* __restrict__ A, const float* __restrict__ W,
    const float* __restrict__ bias, float* __restrict__ C,
    int M, int N, int K, long long ldc) {
  __shared__ float As[64 * 17];
  __shared__ float Ws[128 * 17];

  const int tid = threadIdx.x;
  const int wid = tid >> 5;
  const int lane = tid & 31;
  const int hi = lane >> 4;   // which K-half of the fragment this lane holds
  const int l = lane & 15;    // row (A) / col (B) within 16
  const int mw = wid & 3;     // wave's 16-row slice of the 64-row block tile
  const int nw = wid >> 2;    // wave's 64-col slice of the 128-col block tile
  const int m0 = blockIdx.y * 64;
  const int n0 = blockIdx.x * 128;
  const int lrow = tid >> 2;        // cooperative-load row 0..63
  const int lcol = (tid & 3) << 2;  // cooperative-load col {0,4,8,12}

  v8f c0 = {};
  v8f c1 = {};
  v8f c2 = {};
  v8f c3 = {};

  for (int k0 = 0; k0 < K; k0 += 16) {
    const float4 av = *(const float4*)(A + (size_t)(m0 + lrow) * K + k0 + lcol);
    const float4 wv0 =
        *(const float4*)(W + (size_t)(n0 + lrow) * K + k0 + lcol);
    const float4 wv1 =
        *(const float4*)(W + (size_t)(n0 + 64 + lrow) * K + k0 + lcol);
    As[lrow * 17 + lcol + 0] = av.x;
    As[lrow * 17 + lcol + 1] = av.y;
    As[lrow * 17 + lcol + 2] = av.z;
    As[lrow * 17 + lcol + 3] = av.w;
    Ws[lrow * 17 + lcol + 0] = wv0.x;
    Ws[lrow * 17 + lcol + 1] = wv0.y;
    Ws[lrow * 17 + lcol + 2] = wv0.z;
    Ws[lrow * 17 + lcol + 3] = wv0.w;
    Ws[(64 + lrow) * 17 + lcol + 0] = wv1.x;
    Ws[(64 + lrow) * 17 + lcol + 1] = wv1.y;
    Ws[(64 + lrow) * 17 + lcol + 2] = wv1.z;
    Ws[(64 + lrow) * 17 + lcol + 3] = wv1.w;
    __syncthreads();

#pragma unroll
    for (int kk = 0; kk < 16; kk += 4) {
      const int ka = kk + 2 * hi;  // lanes 0-15: K{kk,kk+1}; 16-31: K{kk+2,kk+3}
      v2f a, b0, b1, b2, b3;
      a.x = As[(mw * 16 + l) * 17 + ka];
      a.y = As[(mw * 16 + l) * 17 + ka + 1];
      const int wb = (nw * 64 + l) * 17 + ka;
      b0.x = Ws[wb + 0 * 16 * 17];
      b0.y = Ws[wb + 0 * 16 * 17 + 1];
      b1.x = Ws[wb + 1 * 16 * 17];
      b1.y = Ws[wb + 1 * 16 * 17 + 1];
      b2.x = Ws[wb + 2 * 16 * 17];
      b2.y = Ws[wb + 2 * 16 * 17 + 1];
      b3.x = Ws[wb + 3 * 16 * 17];
      b3.y = Ws[wb + 3 * 16 * 17 + 1];
      c0 = __builtin_amdgcn_wmma_f32_16x16x4_f32(false, a, false, b0,
                                                 (short)0, c0, false, false);
      c1 = __builtin_amdgcn_wmma_f32_16x16x4_f32(false, a, false, b1,
                                                 (short)0, c1, false, false);
      c2 = __builtin_amdgcn_wmma_f32_16x16x4_f32(false, a, false, b2,
                                                 (short)0, c2, false, false);
      c3 = __builtin_amdgcn_wmma_f32_16x16x4_f32(false, a, false, b3,
                                                 (short)0, c3, false, false);
    }
    __syncthreads();
  }

  // C/D layout: VGPR i -> M = i (lanes 0-15) or i+8 (lanes 16-31), N = l
  const int row0 = m0 + mw * 16 + hi * 8;
  const int col0 = n0 + nw * 64 + l;
  const float bb0 = bias[col0];
  const float bb1 = bias[col0 + 16];
  const float bb2 = bias[col0 + 32];
  const float bb3 = bias[col0 + 48];
#pragma unroll
  for (int i = 0; i < 8; ++i) {
    const long long r = row0 + i;
    C[r * ldc + col0] = c0[i] + bb0;
    C[r * ldc + col0 + 16] = c1[i] + bb1;
    C[r * ldc + col0 + 32] = c2[i] + bb2;
    C[r * ldc + col0 + 48] = c3[i] + bb3;
  }
}

// Embedding gather with padding_idx=0 semantics; writes e into x[:, 0:H].
__global__ void embed_kernel(const float* __restrict__ emb,
                             const int* __restrict__ target,
                             float* __restrict__ x, int t) {
  const int b = blockIdx.x;
  const int tok = (t == 0) ? SOS_ : target[b * T_ + (t - 1)];
  const float s = (tok == 0) ? 0.0f : 1.0f;
  const float* er = emb + (size_t)tok * H_;
  float* xr = x + (size_t)b * (2 * H_);
  for (int h = threadIdx.x; h < H_; h += blockDim.x) xr[h] = er[h] * s;
}

// Bahdanau attention for one batch row: scores -> softmax -> context.
// Writes ctx into x[:, H:2H].
__global__ __launch_bounds__(256) void attn_kernel(
    const float* __restrict__ q, const float* __restrict__ Uk,
    const float* __restrict__ enc, const float* __restrict__ Va_w,
    const float* __restrict__ Va_b, float* __restrict__ x) {
  __shared__ float sc[64];
  __shared__ float qsh[H_];
  const int b = blockIdx.x;
  const int tid = threadIdx.x;
  const int wid = tid >> 5;
  const int lane = tid & 31;

  for (int h = tid; h < H_; h += 256) qsh[h] = q[(size_t)b * H_ + h];
  __syncthreads();

  // scores[s] = Va . tanh(q + Uk[b,s]) + Va_b  (one wave per s)
  for (int s = wid; s < S_; s += 8) {
    const float* ukr = Uk + ((size_t)b * S_ + s) * H_;
    float acc = 0.0f;
    for (int h = lane; h < H_; h += 32)
      acc += tanhf(qsh[h] + ukr[h]) * Va_w[h];
    for (int off = 16; off > 0; off >>= 1) acc += __shfl_xor(acc, off, 32);
    if (lane == 0) sc[s] = acc + Va_b[0];
  }
  __syncthreads();

  // softmax over S_ (wave 0)
  if (wid == 0) {
    float m = -3.4e38f;
    for (int s = lane; s < S_; s += 32) m = fmaxf(m, sc[s]);
    for (int off = 16; off > 0; off >>= 1) m = fmaxf(m, __shfl_xor(m, off, 32));
    float sum = 0.0f;
    for (int s = lane; s < S_; s += 32) {
      const float e = expf(sc[s] - m);
      sc[s] = e;
      sum += e;
    }
    for (int off = 16; off > 0; off >>= 1) sum += __shfl_xor(sum, off, 32);
    const float inv = 1.0f / sum;
    for (int s = lane; s < S_; s += 32) sc[s] *= inv;
  }
  __syncthreads();

  // ctx[h] = sum_s w[s] * enc[b,s,h]
  float* ctx = x + (size_t)b * (2 * H_) + H_;
  for (int h = tid; h < H_; h += 256) {
    const float* er = enc + (size_t)b * S_ * H_ + h;
    float acc = 0.0f;
    for (int s = 0; s < S_; ++s) acc += sc[s] * er[(size_t)s * H_];
    ctx[h] = acc;
  }
}

// GRU gate combine: h_new = (1-z)*n + z*h
__global__ void gru_combine(const float* __restrict__ gi,
                            const float* __restrict__ gh,
                            const float* __restrict__ h,
                            float* __restrict__ hn) {
  const int b = blockIdx.x;
  const float* gib = gi + (size_t)b * 3 * H_;
  const float* ghb = gh + (size_t)b * 3 * H_;
  for (int j = threadIdx.x; j < H_; j += blockDim.x) {
    const float ir = gib[j], iz = gib[H_ + j], in_ = gib[2 * H_ + j];
    const float hr = ghb[j], hz = ghb[H_ + j], hn_ = ghb[2 * H_ + j];
    const float r = 1.0f / (1.0f + expf(-(ir + hr)));
    const float z = 1.0f / (1.0f + expf(-(iz + hz)));
    const float n = tanhf(in_ + r * hn_);
    hn[(size_t)b * H_ + j] = (1.0f - z) * n + z * h[(size_t)b * H_ + j];
  }
}

// In-place log_softmax over one V_-length row per block.
__global__ __launch_bounds__(512) void logsoftmax_kernel(float* __restrict__ base) {
  __shared__ float red[16];
  float* row = base + (size_t)blockIdx.x * ((size_t)T_ * V_);
  const int tid = threadIdx.x;

  float m = -3.4e38f;
  for (int v = tid; v < V_; v += 512) m = fmaxf(m, row[v]);
  for (int off = 16; off > 0; off >>= 1) m = fmaxf(m, __shfl_xor(m, off, 32));
  if ((tid & 31) == 0) red[tid >> 5] = m;
  __syncthreads();
  float mb = red[0];
#pragma unroll
  for (int i = 1; i < 16; ++i) mb = fmaxf(mb, red[i]);
  __syncthreads();

  float sum = 0.0f;
  for (int v = tid; v < V_; v += 512) sum += expf(row[v] - mb);
  for (int off = 16; off > 0; off >>= 1) sum += __shfl_xor(sum, off, 32);
  if ((tid & 31) == 0) red[tid >> 5] = sum;
  __syncthreads();
  float tot = 0.0f;
#pragma unroll
  for (int i = 0; i < 16; ++i) tot += red[i];

  const float lse = mb + logf(tot);
  for (int v = tid; v < V_; v += 512) row[v] -= lse;
}

extern "C" void kernel_launch(void* const* d_in, const int* in_sizes, int n_in,
                              void* d_out, int out_size, void* d_ws, size_t ws_size,
                              hipStream_t stream) {
  const float* enc   = (const float*)d_in[0];   // (B,S,H)
  const float* h0    = (const float*)d_in[1];   // (1,B,H)
  const float* emb   = (const float*)d_in[2];   // (V,H)
  const float* Wa_w  = (const float*)d_in[3];   // (H,H)
  const float* Wa_b  = (const float*)d_in[4];   // (H)
  const float* Ua_w  = (const float*)d_in[5];   // (H,H)
  const float* Ua_b  = (const float*)d_in[6];   // (H)
  const float* Va_w  = (const float*)d_in[7];   // (1,H)
  const float* Va_b  = (const float*)d_in[8];   // (1)
  const float* W_ih  = (const float*)d_in[9];   // (3H,2H)
  const float* W_hh  = (const float*)d_in[10];  // (3H,H)
  const float* b_ih  = (const float*)d_in[11];  // (3H)
  const float* b_hh  = (const float*)d_in[12];  // (3H)
  const float* out_w = (const float*)d_in[13];  // (V,H)
  const float* out_b = (const float*)d_in[14];  // (V)
  const int* target  = (const int*)d_in[15];    // (B,T)
  float* out = (float*)d_out;                   // (B,T,V)

  float* ws = (float*)d_ws;
  size_t o = 0;
  float* Uk = ws + o; o += (size_t)B_ * S_ * H_;  // 3.93M floats
  float* x  = ws + o; o += (size_t)B_ * 2 * H_;   // [e | ctx]
  float* q  = ws + o; o += (size_t)B_ * H_;
  float* gi = ws + o; o += (size_t)B_ * 3 * H_;
  float* gh = ws + o; o += (size_t)B_ * 3 * H_;
  float* hA = ws + o; o += (size_t)B_ * H_;
  float* hB = ws + o;

  // h0 -> working hidden-state buffer (capture-safe D2D)
  hipMemcpyAsync(hA, h0, (size_t)B_ * H_ * sizeof(float),
                 hipMemcpyDeviceToDevice, stream);

  // Precompute Uk = enc @ Ua_w^T + Ua_b   (M=B*S=3840, N=H, K=H)
  gemm_atb<<<dim3(H_ / 128, (B_ * S_) / 64), 256, 0, stream>>>(
      enc, Ua_w, Ua_b, Uk, B_ * S_, H_, H_, (long long)H_);

  float* hc = hA;
  float* hn = hB;
  for (int t = 0; t < T_; ++t) {
    embed_kernel<<<B_, 256, 0, stream>>>(emb, target, x, t);
    // q = h @ Wa_w^T + Wa_b
    gemm_atb<<<dim3(H_ / 128, 1), 256, 0, stream>>>(
        hc, Wa_w, Wa_b, q, B_, H_, H_, (long long)H_);
    attn_kernel<<<B_, 256, 0, stream>>>(q, Uk, enc, Va_w, Va_b, x);
    // gi = [e|ctx] @ W_ih^T + b_ih  (K = 2H)
    gemm_atb<<<dim3(3 * H_ / 128, 1), 256, 0, stream>>>(
        x, W_ih, b_ih, gi, B_, 3 * H_, 2 * H_, (long long)(3 * H_));
    // gh = h @ W_hh^T + b_hh
    gemm_atb<<<dim3(3 * H_ / 128, 1), 256, 0, stream>>>(
        hc, W_hh, b_hh, gh, B_, 3 * H_, H_, (long long)(3 * H_));
    gru_combine<<<B_, 256, 0, stream>>>(gi, gh, hc, hn);
    // logits directly into d_out rows: row b at out + b*T*V + t*V
    gemm_atb<<<dim3(V_ / 128, 1), 256, 0, stream>>>(
        hn, out_w, out_b, out + (size_t)t * V_, B_, V_, H_,
        (long long)T_ * V_);
    logsoftmax_kernel<<<B_, 512, 0, stream>>>(out + (size_t)t * V_);
    float* tmp = hc; hc = hn; hn = tmp;
  }
}